// MultiChannelGRU_59700045414884
// MI455X (gfx1250) — compile-verified
//
#include <hip/hip_runtime.h>
#include <hip/hip_bf16.h>
#include <math.h>

// ---------------------------------------------------------------------------
// MultiChannelGRU on gfx1250 using native FP32 WMMA (v_wmma_f32_16x16x4_f32).
// B=32, C=8 -> NSEQ=256 sequences, T=64, F=256, H=512, 3H=1536.
// Layer recurrences run as single persistent kernels with a grid barrier
// between time steps (128 co-resident workgroups), eliminating 128
// sequential kernel launches.
// ---------------------------------------------------------------------------

typedef float v2f __attribute__((ext_vector_type(2)));
typedef float v8f __attribute__((ext_vector_type(8)));

#define NSEQ 256
#define TLEN 64
#define FDIM 256
#define HDIM 512
#define G3   1536   // 3*H

__device__ __forceinline__ v8f wmma4(v2f a, v2f b, v8f c) {
    // D(16x16,f32) = A(16x4,f32) * B(4x16,f32) + C
    return __builtin_amdgcn_wmma_f32_16x16x4_f32(
        false, a, false, b, (short)0, c, false, false);
}

__device__ __forceinline__ float sigmoidf_(float v) {
    return 1.0f / (1.0f + expf(-v));
}

// Generation-counter grid barrier. bar[0]=arrive count, bar[1]=generation.
__device__ __forceinline__ void grid_sync(unsigned* bar, unsigned nblk) {
    __syncthreads();
    __threadfence();   // release: make this block's h writes agent-visible
    if (threadIdx.x == 0) {
        unsigned gen = __hip_atomic_load(&bar[1], __ATOMIC_RELAXED,
                                         __HIP_MEMORY_SCOPE_AGENT);
        unsigned arrived = __hip_atomic_fetch_add(&bar[0], 1u, __ATOMIC_ACQ_REL,
                                                  __HIP_MEMORY_SCOPE_AGENT) + 1u;
        if (arrived == nblk) {
            __hip_atomic_store(&bar[0], 0u, __ATOMIC_RELAXED,
                               __HIP_MEMORY_SCOPE_AGENT);
            __hip_atomic_fetch_add(&bar[1], 1u, __ATOMIC_RELEASE,
                                   __HIP_MEMORY_SCOPE_AGENT);
        } else {
            while (__hip_atomic_load(&bar[1], __ATOMIC_ACQUIRE,
                                     __HIP_MEMORY_SCOPE_AGENT) == gen) {
                __builtin_amdgcn_s_sleep(2);
            }
        }
    }
    __syncthreads();
    __threadfence();   // acquire: discard stale cached h lines
}

__global__ void init_bar_kernel(unsigned* bar) {
    if (threadIdx.x < 2) bar[threadIdx.x] = 0u;
}

// ---------------------------------------------------------------------------
// D[M,1536] = X[M,K] @ W[1536,K]^T + bias[1536]
// 2x2 register tiling per wave: 2 A-frags + 2 B-frags feed 4 WMMAs/K-chunk.
// grid = (1536/128, M/32), block = 128 (4 waves, wave w owns 32-col slice).
// Fragment mapping (wave32, v_wmma_f32_16x16x4_f32):
//   A: lane L holds X[m + L%16][k0 + 2*(L/16) + {0,1}]
//   B: lane L holds W[n + L%16][k0 + 2*(L/16) + {0,1}]   (B[k][n] = W[n][k])
//   C: VGPR g, lane L -> row m + g + 8*(L/16), col n + L%16
// ---------------------------------------------------------------------------
__global__ void gemm_bias_kernel(const float* __restrict__ X,
                                 const float* __restrict__ W,
                                 const float* __restrict__ bias,
                                 float* __restrict__ D, int K) {
    const int lane = threadIdx.x & 31;
    const int wv   = threadIdx.x >> 5;
    const int half = lane >> 4;
    const int l16  = lane & 15;
    const int m0 = blockIdx.y << 5;                  // 2 m-tiles: m0, m0+16
    const int n0 = (blockIdx.x << 7) + (wv << 5);    // 2 n-tiles: n0, n0+16

    const float* x0 = X + (size_t)(m0 + l16) * K + 2 * half;
    const float* x1 = x0 + (size_t)16 * K;
    const float* w0 = W + (size_t)(n0 + l16) * K + 2 * half;
    const float* w1 = w0 + (size_t)16 * K;

    v8f c00 = {}, c01 = {}, c10 = {}, c11 = {};
#pragma unroll 4
    for (int k0 = 0; k0 < K; k0 += 4) {
        v2f a0 = *reinterpret_cast<const v2f*>(x0 + k0);
        v2f a1 = *reinterpret_cast<const v2f*>(x1 + k0);
        v2f b0 = *reinterpret_cast<const v2f*>(w0 + k0);
        v2f b1 = *reinterpret_cast<const v2f*>(w1 + k0);
        c00 = wmma4(a0, b0, c00);
        c01 = wmma4(a0, b1, c01);
        c10 = wmma4(a1, b0, c10);
        c11 = wmma4(a1, b1, c11);
    }

    const int col0 = n0 + l16;
    const int col1 = col0 + 16;
    const float bs0 = bias[col0];
    const float bs1 = bias[col1];
#pragma unroll
    for (int g = 0; g < 8; ++g) {
        const int row0 = m0 + g + 8 * half;
        const int row1 = row0 + 16;
        D[(size_t)row0 * G3 + col0] = c00[g] + bs0;
        D[(size_t)row0 * G3 + col1] = c01[g] + bs1;
        D[(size_t)row1 * G3 + col0] = c10[g] + bs0;
        D[(size_t)row1 * G3 + col1] = c11[g] + bs1;
    }
}

// ---------------------------------------------------------------------------
// Persistent full-layer GRU recurrence. One launch covers all T=64 steps,
// with a grid barrier between steps. grid = (H/64, NSEQ/16) = 128 blocks,
// block = 128 threads (4 waves). Each wave owns one 16x16 (seq x hidden)
// tile and computes the r/z/n gh tiles sharing a single A fragment per
// K-chunk (3 independent WMMA chains). h0 == 0, so step 0 skips the matmul.
// h ping-pongs between hA/hB; final h (t=T-1) lands in hA.
// ---------------------------------------------------------------------------
__global__ void gru_layer_kernel(const float* __restrict__ Whh,
                                 const float* __restrict__ bhh,
                                 const float* __restrict__ gi,  // [NSEQ*T, 1536]
                                 float* __restrict__ hA,
                                 float* __restrict__ hB,
                                 float* __restrict__ y,         // [NSEQ*T, H] or null
                                 unsigned* __restrict__ bar) {
    const int lane = threadIdx.x & 31;
    const int wv   = threadIdx.x >> 5;
    const int half = lane >> 4;
    const int l16  = lane & 15;
    const int m0 = blockIdx.y << 4;                  // sequence-row tile
    const int n0 = (blockIdx.x << 6) + (wv << 4);    // hidden-col tile

    const float* wr = Whh + (size_t)(n0 + l16) * HDIM + 2 * half;
    const float* wz = wr + (size_t)HDIM * HDIM;
    const float* wn = wr + (size_t)(2 * HDIM) * HDIM;

    const int col = n0 + l16;
    const float br = bhh[col];
    const float bz = bhh[col + HDIM];
    const float bn = bhh[col + 2 * HDIM];
    const unsigned nblk = gridDim.x * gridDim.y;     // 128

    for (int t = 0; t < TLEN; ++t) {
        const float* hp = (t & 1) ? hB : hA;
        float*       hn = (t & 1) ? hA : hB;

        v8f cr = {}, cz = {}, cn = {};
        if (t > 0) {   // grid-uniform branch: EXEC all-ones inside
            const float* arow = hp + (size_t)(m0 + l16) * HDIM + 2 * half;
#pragma unroll 4
            for (int k0 = 0; k0 < HDIM; k0 += 4) {
                v2f a = *reinterpret_cast<const v2f*>(arow + k0);
                cr = wmma4(a, *reinterpret_cast<const v2f*>(wr + k0), cr);
                cz = wmma4(a, *reinterpret_cast<const v2f*>(wz + k0), cz);
                cn = wmma4(a, *reinterpret_cast<const v2f*>(wn + k0), cn);
            }
        }

#pragma unroll
        for (int g = 0; g < 8; ++g) {
            const int row = m0 + g + 8 * half;
            const size_t gb = ((size_t)row * TLEN + t) * G3 + col;
            const float r  = sigmoidf_(gi[gb]            + cr[g] + br);
            const float z  = sigmoidf_(gi[gb + HDIM]     + cz[g] + bz);
            const float nv = tanhf   (gi[gb + 2 * HDIM] + r * (cn[g] + bn));
            const float hpv = (t > 0) ? hp[(size_t)row * HDIM + col] : 0.0f;
            const float hv = (1.0f - z) * nv + z * hpv;
            hn[(size_t)row * HDIM + col] = hv;
            if (y) y[((size_t)row * TLEN + t) * HDIM + col] = hv;
        }

        if (t != TLEN - 1) grid_sync(bar, nblk);
    }
}

// ---------------------------------------------------------------------------
// out[b] = mean_c( h[b*8+c] . fcW + fcb ),  b in [0,32)
// ---------------------------------------------------------------------------
__global__ void fc_mean_kernel(const float* __restrict__ h,
                               const float* __restrict__ fcW,
                               const float* __restrict__ fcb,
                               float* __restrict__ out) {
    __shared__ float red[256];
    const int b = blockIdx.x;
    const int tid = threadIdx.x;
    float s = 0.0f;
    for (int c = 0; c < 8; ++c) {
        const float* hr = h + (size_t)(b * 8 + c) * HDIM;
        for (int k = tid; k < HDIM; k += 256) s += hr[k] * fcW[k];
    }
    red[tid] = s;
    __syncthreads();
    for (int o = 128; o > 0; o >>= 1) {
        if (tid < o) red[tid] += red[tid + o];
        __syncthreads();
    }
    if (tid == 0) out[b] = red[0] * 0.125f + fcb[0];
}

// ---------------------------------------------------------------------------
// Host orchestration: 6 graph nodes total (vs 134 before).
// Workspace (floats unless noted, 256B-aligned by construction):
//   gi  : 16384*1536  (100 MB, reused by both layers)
//   y0  : 16384*512   ( 32 MB)
//   hA  : 256*512
//   hB  : 256*512
//   bar : 2 x u32     (grid-barrier state, re-zeroed every call)
// ---------------------------------------------------------------------------
extern "C" void kernel_launch(void* const* d_in, const int* in_sizes, int n_in,
                              void* d_out, int out_size, void* d_ws, size_t ws_size,
                              hipStream_t stream) {
    const float* x    = (const float*)d_in[0];
    const float* Wih0 = (const float*)d_in[1];
    const float* Whh0 = (const float*)d_in[2];
    const float* bih0 = (const float*)d_in[3];
    const float* bhh0 = (const float*)d_in[4];
    const float* Wih1 = (const float*)d_in[5];
    const float* Whh1 = (const float*)d_in[6];
    const float* bih1 = (const float*)d_in[7];
    const float* bhh1 = (const float*)d_in[8];
    const float* fcW  = (const float*)d_in[9];
    const float* fcb  = (const float*)d_in[10];
    float* out = (float*)d_out;

    float* gi = (float*)d_ws;
    float* y0 = gi + (size_t)NSEQ * TLEN * G3;
    float* hA = y0 + (size_t)NSEQ * TLEN * HDIM;
    float* hB = hA + (size_t)NSEQ * HDIM;
    unsigned* bar = (unsigned*)(hB + (size_t)NSEQ * HDIM);

    const dim3 gemmGrid(G3 / 128, (NSEQ * TLEN) / 32);  // (12, 512)
    const dim3 layerGrid(HDIM / 64, NSEQ / 16);         // (8, 16) = 128 blocks

    init_bar_kernel<<<1, 32, 0, stream>>>(bar);

    // ---- Layer 0 ----
    gemm_bias_kernel<<<gemmGrid, 128, 0, stream>>>(x, Wih0, bih0, gi, FDIM);
    gru_layer_kernel<<<layerGrid, 128, 0, stream>>>(Whh0, bhh0, gi, hA, hB, y0, bar);

    // ---- Layer 1 (reuse gi buffer; only final hidden state needed) ----
    gemm_bias_kernel<<<gemmGrid, 128, 0, stream>>>(y0, Wih1, bih1, gi, HDIM);
    gru_layer_kernel<<<layerGrid, 128, 0, stream>>>(Whh1, bhh1, gi, hA, hB, nullptr, bar);

    // ---- FC + channel mean (final h is in hA after even #steps) ----
    fc_mean_kernel<<<32, 256, 0, stream>>>(hA, fcW, fcb, out);
}